// SimilarityGuidedSampling_84542136254600
// MI455X (gfx1250) — compile-verified
//
#include <hip/hip_runtime.h>

typedef __attribute__((ext_vector_type(2))) float v2f;
typedef __attribute__((ext_vector_type(8))) float v8f;

#define BB   8
#define CC   64
#define TT   32
#define HHWW 3136   // 56*56
#define NB   4
#define HID  32
#define EMB  8

// ---------------------------------------------------------------------------
// Kernel 1: pooled[b,c,t] = mean over H*W of x[b,c,t,:,:]
// One block per (b,c,t); coalesced stride-256 reads, wave32 shuffle reduce.
// ---------------------------------------------------------------------------
__global__ void __launch_bounds__(256)
sgs_pool_kernel(const float* __restrict__ x, float* __restrict__ pooled) {
    const int bct = blockIdx.x;                 // (b*C + c)*T + t
    const float* xp = x + (size_t)bct * HHWW;
    float s = 0.f;
    for (int i = threadIdx.x; i < HHWW; i += 256) s += xp[i];
    #pragma unroll
    for (int off = 16; off > 0; off >>= 1) s += __shfl_xor(s, off, 32);
    __shared__ float smem[8];
    const int wave = threadIdx.x >> 5;
    const int lane = threadIdx.x & 31;
    if (lane == 0) smem[wave] = s;
    __syncthreads();
    if (threadIdx.x == 0) {
        float tot = 0.f;
        #pragma unroll
        for (int w = 0; w < 8; ++w) tot += smem[w];
        pooled[bct] = tot * (1.0f / (float)HHWW);
    }
}

// ---------------------------------------------------------------------------
// Kernel 2: tiny MLP + hswish + norms + per-batch gamma -> coeff[b,t,n].
// One wave (32 threads == T) per batch; min/max via wave32 shuffles.
// ---------------------------------------------------------------------------
__global__ void __launch_bounds__(32)
sgs_coeff_kernel(const float* __restrict__ pooled,
                 const float* __restrict__ w1, const float* __restrict__ b1,
                 const float* __restrict__ w2, const float* __restrict__ b2,
                 const int*   __restrict__ num_bins_p,
                 float* __restrict__ coeff) {
    const int b = blockIdx.x;   // [0, B)
    const int t = threadIdx.x;  // [0, 32) == T

    // h = w1 @ pooled[b,:,t] + b1
    float h[HID];
    #pragma unroll
    for (int o = 0; o < HID; ++o) h[o] = b1[o];
    for (int c = 0; c < CC; ++c) {
        const float p = pooled[(b * CC + c) * TT + t];   // coalesced across lanes
        #pragma unroll
        for (int o = 0; o < HID; ++o) h[o] += w1[o * CC + c] * p;
    }
    // hswish
    #pragma unroll
    for (int o = 0; o < HID; ++o) {
        float r = fminf(fmaxf(h[o] + 3.f, 0.f), 6.f);
        h[o] = h[o] * r * (1.f / 6.f);
    }
    // embds + squared norm
    float norm = 0.f;
    #pragma unroll
    for (int j = 0; j < EMB; ++j) {
        float e = b2[j];
        #pragma unroll
        for (int o = 0; o < HID; ++o) e += w2[j * HID + o] * h[o];
        norm += e * e;
    }
    // wave-wide min/max over the T=32 lanes
    float mx = norm, mn = norm;
    #pragma unroll
    for (int off = 16; off > 0; off >>= 1) {
        mx = fmaxf(mx, __shfl_xor(mx, off, 32));
        mn = fminf(mn, __shfl_xor(mn, off, 32));
    }
    const int nb = num_bins_p[0];                    // == 4 in this dispatch
    const float gamma = 0.5f * (mx - mn) / (float)nb;
    #pragma unroll
    for (int n = 0; n < NB; ++n) {
        const float center = gamma * (2.f * (float)n + 1.f);
        const float cf = fmaxf(1.f - fabsf(norm - center) / gamma, 0.f);
        coeff[(b * TT + t) * NB + n] = cf;
    }
}

// ---------------------------------------------------------------------------
// Kernel 3: out[b,c,n,hw] = sum_t coeff[b,t,n] * x[b,c,t,hw]
// via V_WMMA_F32_16X16X4_F32, K=4 per issue, 8 issues cover T=32.
//   A (16x4): M=bin rows (4 live), lane L: M=L%16, vgpr j: K=(L<16?0:2)+j
//   B (4x16): N=hw cols,           lane L: N=L%16, vgpr j: K=(L<16?0:2)+j
//   D: lanes 0-15, vgpr n (n<4) hold out[n, hw_base + lane]
// One wave per 16-wide hw tile of one (b,c); tile choice is wave-uniform so
// EXEC is all-1s at every WMMA.
// ---------------------------------------------------------------------------
__global__ void __launch_bounds__(256)
sgs_wsum_kernel(const float* __restrict__ x, const float* __restrict__ coeff,
                float* __restrict__ out) {
    const int bc   = blockIdx.x;            // [0, B*C)
    const int b    = bc / CC;
    const int wave = threadIdx.x >> 5;
    const int lane = threadIdx.x & 31;
    const int tile = blockIdx.y * 8 + wave; // 16-wide hw tile index
    if (tile >= HHWW / 16) return;          // uniform per wave

    const int m     = lane & 15;            // M row (A) / N col (B,D)
    const int khalf = (lane >> 4) << 1;     // 0 (lanes 0-15) or 2 (lanes 16-31)

    // A fragments: coeff tiles, constant across all (c, hw) of this batch.
    v2f a[8];
    const float* cb = coeff + (size_t)b * (TT * NB);
    #pragma unroll
    for (int tq = 0; tq < 8; ++tq) {
        const int t0 = tq * 4 + khalf;
        a[tq][0] = (m < NB) ? cb[t0 * NB + m]       : 0.f;
        a[tq][1] = (m < NB) ? cb[(t0 + 1) * NB + m] : 0.f;
    }

    const int hw = tile * 16 + m;
    const float* xb = x + (size_t)bc * (TT * HHWW) + hw;

    v8f acc = {};
    #pragma unroll
    for (int tq = 0; tq < 8; ++tq) {
        const int t0 = tq * 4 + khalf;
        v2f bf;
        bf[0] = xb[(size_t)t0 * HHWW];        // 64B-contiguous per half-wave
        bf[1] = xb[(size_t)(t0 + 1) * HHWW];
        acc = __builtin_amdgcn_wmma_f32_16x16x4_f32(
            /*neg_a=*/false, a[tq], /*neg_b=*/false, bf,
            /*c_mod=*/(short)0, acc, /*reuse_a=*/false, /*reuse_b=*/false);
    }

    if (lane < 16) {   // rows 0..3 of D live in vgprs 0..3 of the low half-wave
        float* ob = out + (size_t)bc * (NB * HHWW) + hw;
        #pragma unroll
        for (int n = 0; n < NB; ++n) ob[(size_t)n * HHWW] = acc[n];
    }
}

// ---------------------------------------------------------------------------
extern "C" void kernel_launch(void* const* d_in, const int* in_sizes, int n_in,
                              void* d_out, int out_size, void* d_ws, size_t ws_size,
                              hipStream_t stream) {
    const float* x        = (const float*)d_in[0];
    const float* w1       = (const float*)d_in[1];
    const float* b1       = (const float*)d_in[2];
    const float* w2       = (const float*)d_in[3];
    const float* b2       = (const float*)d_in[4];
    const int*   num_bins = (const int*)  d_in[5];
    float* out = (float*)d_out;

    float* pooled = (float*)d_ws;                 // B*C*T   = 16384 floats
    float* coeff  = pooled + BB * CC * TT;        // B*T*NB  = 1024 floats

    sgs_pool_kernel<<<dim3(BB * CC * TT), 256, 0, stream>>>(x, pooled);
    sgs_coeff_kernel<<<dim3(BB), 32, 0, stream>>>(pooled, w1, b1, w2, b2,
                                                  num_bins, coeff);
    const int tiles = HHWW / 16;                  // 196
    sgs_wsum_kernel<<<dim3(BB * CC, (tiles + 7) / 8), 256, 0, stream>>>(x, coeff, out);
}